// GNN_84559316123938
// MI455X (gfx1250) — compile-verified
//
#include <hip/hip_runtime.h>

#define N_NODES 100000
#define F 128
#define TILE_M 128   // 8 waves * 16 rows per block

typedef __attribute__((ext_vector_type(2))) float v2f;
typedef __attribute__((ext_vector_type(4))) float v4f;
typedef __attribute__((ext_vector_type(8))) float v8f;

// ---------------------------------------------------------------------------
// Scatter-add: agg[dst[e]] += x[src[e]]   (one wave32 per edge, float4 lanes)
// ---------------------------------------------------------------------------
__global__ __launch_bounds__(256) void gnn_scatter_add(
    const float* __restrict__ x, const int* __restrict__ src,
    const int* __restrict__ dst, float* __restrict__ agg, int E)
{
    int e = blockIdx.x * 8 + (threadIdx.x >> 5);
    if (e >= E) return;
    int lane = threadIdx.x & 31;
    int s = src[e];
    int d = dst[e];
    const v4f* xs = (const v4f*)(x + (size_t)s * F);
    v4f v = xs[lane];
    float* p = agg + (size_t)d * F + lane * 4;
    atomicAdd(p + 0, v.x);
    atomicAdd(p + 1, v.y);
    atomicAdd(p + 2, v.z);
    atomicAdd(p + 3, v.w);
}

// ---------------------------------------------------------------------------
// Fused GraphConv GEMM:  out = epi( agg @ Wr + b + xin @ Wroot )
//   mode bit0: relu, bit1: add residual (xin)
// [agg|xin] (N x 256) @ [Wr;Wroot] (256 x 128) via V_WMMA_F32_16X16X4_F32.
// B K-panel staged in LDS in K-pair-interleaved layout so each lane's B
// fragment is one aligned ds_load_b64 (no repacking movs in the hot loop).
// ---------------------------------------------------------------------------
__global__ __launch_bounds__(256) void gnn_gemm_fused(
    const float* __restrict__ agg, const float* __restrict__ xin,
    const float* __restrict__ Wr,  const float* __restrict__ Wroot,
    const float* __restrict__ bias, float* __restrict__ out,
    int n_nodes, int mode)
{
    // ldsB[p*128 + n] = { W[2p][n], W[2p+1][n] }  for panel k-pairs p=0..7
    __shared__ v2f ldsB[8 * F];             // 8 KB

    const int tid  = threadIdx.x;
    const int wave = tid >> 5;
    const int lane = tid & 31;
    const int half = lane >> 4;             // 0: K pair {0,1}, 1: K pair {2,3}
    const int l16  = lane & 15;

    const int mbase = blockIdx.x * TILE_M + wave * 16;
    int arow = mbase + l16;                 // A row for this lane (A layout)
    if (arow >= n_nodes) arow = n_nodes - 1;  // clamp: EXEC must stay all-1s

    v8f acc[8];
#pragma unroll
    for (int nt = 0; nt < 8; ++nt) acc[nt] = (v8f){};

    // K = 256 total: k0 in [0,128) -> agg @ Wr ; k0 in [128,256) -> xin @ Wroot
    for (int kp = 0; kp < 16; ++kp) {
        const int k0 = kp * 16;
        const float* Wsrc = (k0 < F) ? (Wr + (size_t)k0 * F)
                                     : (Wroot + (size_t)(k0 - F) * F);
        // cooperative interleaved stage: 512 (p, n-pair) items, 2 per thread
#pragma unroll
        for (int i = 0; i < 2; ++i) {
            const int q  = tid + 256 * i;       // 0..511
            const int p  = q >> 6;              // k-pair 0..7
            const int n2 = (q & 63) * 2;        // n base (even)
            v2f r0 = *(const v2f*)&Wsrc[(size_t)(p * 2)     * F + n2];
            v2f r1 = *(const v2f*)&Wsrc[(size_t)(p * 2 + 1) * F + n2];
            ldsB[p * F + n2]     = (v2f){r0.x, r1.x};
            ldsB[p * F + n2 + 1] = (v2f){r0.y, r1.y};
        }
        __syncthreads();

        const float* Asrc = (k0 < F) ? agg : xin;
        const int kloc = k0 & (F - 1);
#pragma unroll
        for (int ks = 0; ks < 4; ++ks) {
            // A fragment: lane holds A[row][kk], A[row][kk+1]
            const int kk = kloc + ks * 4 + half * 2;
            v2f a = *(const v2f*)(Asrc + (size_t)arow * F + kk);
            const int p = ks * 2 + half;        // k-pair within panel
#pragma unroll
            for (int nt = 0; nt < 8; ++nt) {
                const int n = nt * 16 + l16;
                v2f b = ldsB[p * F + n];        // one aligned ds_load_b64
                acc[nt] = __builtin_amdgcn_wmma_f32_16x16x4_f32(
                    /*neg_a=*/false, a, /*neg_b=*/false, b,
                    /*c_mod=*/(short)0, acc[nt],
                    /*reuse_a=*/false, /*reuse_b=*/false);
            }
        }
        __syncthreads();
    }

    // Epilogue: D layout = lanes give N column (l16), VGPR r + half*8 gives M.
#pragma unroll
    for (int nt = 0; nt < 8; ++nt) {
        const int n = nt * 16 + l16;
        const float bv = bias[n];
#pragma unroll
        for (int r = 0; r < 8; ++r) {
            const int row = mbase + half * 8 + r;
            if (row < n_nodes) {
                float v = acc[nt][r] + bv;
                if (mode & 1) v = fmaxf(v, 0.0f);
                if (mode & 2) v += xin[(size_t)row * F + n];
                out[(size_t)row * F + n] = v;
            }
        }
    }
}

// ---------------------------------------------------------------------------
extern "C" void kernel_launch(void* const* d_in, const int* in_sizes, int n_in,
                              void* d_out, int out_size, void* d_ws, size_t ws_size,
                              hipStream_t stream) {
    const float* x     = (const float*)d_in[0];   // [N, 128]
    const int*   ei    = (const int*)d_in[1];     // [2, E]
    const float* Wrel  = (const float*)d_in[2];   // [5, 128, 128]
    const float* brel  = (const float*)d_in[3];   // [5, 128]
    const float* Wroot = (const float*)d_in[4];   // [5, 128, 128]

    const int E = in_sizes[1] / 2;
    const int* src = ei;
    const int* dst = ei + E;

    const size_t NF = (size_t)N_NODES * F;
    float* bufA = (float*)d_ws;
    float* bufB = bufA + NF;
    float* agg  = bufB + NF;

    float* outs[5]  = { bufA, bufB, bufA, bufB, (float*)d_out };
    const int modes[5] = { 1, 3, 3, 3, 2 };   // relu | relu+res | ... | res

    const float* cur = x;
    const int scatter_blocks = (E + 7) / 8;
    const int gemm_blocks = (N_NODES + TILE_M - 1) / TILE_M;

    for (int l = 0; l < 5; ++l) {
        hipMemsetAsync(agg, 0, NF * sizeof(float), stream);
        gnn_scatter_add<<<scatter_blocks, 256, 0, stream>>>(cur, src, dst, agg, E);
        gnn_gemm_fused<<<gemm_blocks, 256, 0, stream>>>(
            agg, cur,
            Wrel + (size_t)l * F * F, Wroot + (size_t)l * F * F,
            brel + (size_t)l * F, outs[l], N_NODES, modes[l]);
        cur = outs[l];
    }
}